// OVDDetector_20813411517099
// MI455X (gfx1250) — compile-verified
//
#include <hip/hip_runtime.h>
#include <hip/hip_bf16.h>

// Match JAX fp32 rounding: no mul+add fma contraction anywhere in this TU.
#pragma clang fp contract(off)

namespace {
constexpr int   kB = 16;
constexpr int   kN = 4096;
constexpr int   kC = 64;
constexpr float kTarget      = 602.0f;
constexpr float kIouThr      = 0.2f;
constexpr float kConfThres   = 0.001f;
constexpr float kBoxConfThr  = 0.01f;
constexpr float kMinBox      = 5.0f;
constexpr float kMaxWh       = 4096.0f;
constexpr float kNegInf      = -1e9f;
}  // namespace

// ---- CDNA5 async global->LDS path (probed via __has_builtin; clean fallback) ----
#if defined(__HIP_DEVICE_COMPILE__) && defined(__has_builtin)
#if __has_builtin(__builtin_amdgcn_global_load_async_to_lds_b128)
#define OVD_ASYNC_LDS 1
#endif
#if __has_builtin(__builtin_amdgcn_s_wait_asynccnt)
#define OVD_HAVE_WAIT_ASYNC_BUILTIN 1
#endif
#endif

#ifdef OVD_ASYNC_LDS
typedef int ovd_v4i __attribute__((ext_vector_type(4)));
typedef __attribute__((address_space(1))) ovd_v4i* ovd_gptr_b128;  // global (AS1)
typedef __attribute__((address_space(3))) ovd_v4i* ovd_lptr_b128;  // LDS (AS3)
#endif

// ============================================================================
// Kernel 1: per-proposal prep — clip boxes, conf/cls = max/argmax(preds*obj),
// validity mask. One thread per (b,n).
// ============================================================================
__global__ __launch_bounds__(256) void k_prep(
    const float* __restrict__ prop, const float* __restrict__ preds,
    const float* __restrict__ obj,
    float* __restrict__ w_box, float* __restrict__ w_conf,
    int* __restrict__ w_cls, unsigned* __restrict__ w_valid) {
  const int g = blockIdx.x * 256 + threadIdx.x;
  if (g >= kB * kN) return;

  const float4 p = ((const float4*)prop)[g];
  const float x1 = fminf(fmaxf(p.x, 0.0f), kTarget);
  const float y1 = fminf(fmaxf(p.y, 0.0f), kTarget);
  const float x2 = fminf(fmaxf(p.z, 0.0f), kTarget);
  const float y2 = fminf(fmaxf(p.w, 0.0f), kTarget);

  const float ob = obj[g];
  const float* pr = preds + (size_t)g * kC;
  __builtin_prefetch(pr, 0, 3);  // gfx1250: global_prefetch_b8

  float best = -3.402823466e38f;
  int bc = 0;
  for (int c = 0; c < kC; ++c) {
    const float v = pr[c] * ob;          // cls_conf = preds * obj
    if (v > best) { best = v; bc = c; }  // strict '>' == JAX first-max argmax
  }

  const bool valid = (ob > kBoxConfThr) && ((x2 - x1) >= kMinBox) &&
                     ((y2 - y1) >= kMinBox) && (best > kConfThres);

  ((float4*)w_box)[g] = make_float4(x1, y1, x2, y2);
  w_conf[g]  = best;
  w_cls[g]   = bc;
  w_valid[g] = valid ? 1u : 0u;
}

// ============================================================================
// Kernel 2: per-batch stable descending sort by conf (bitonic, in LDS) + gather
// sorted boxes / offset-boxes / conf / cls / valid into workspace.
// Key = (ordered_uint(conf) << 32) | ~index  -> ascending sort, read reversed
// gives conf-descending with index-ascending ties (matches stable argsort).
// ============================================================================
__global__ __launch_bounds__(1024) void k_sort(
    const float* __restrict__ w_box, const float* __restrict__ w_conf,
    const int* __restrict__ w_cls, const unsigned* __restrict__ w_valid,
    float* __restrict__ w_sbox, float* __restrict__ w_soff,
    float* __restrict__ w_sconf, float* __restrict__ w_scls,
    unsigned* __restrict__ w_svalid) {
  __shared__ unsigned long long skey[kN];  // 32 KB
  const int b = blockIdx.x;
  const int tid = threadIdx.x;

  for (int s = 0; s < 4; ++s) {
    const int n = tid + s * 1024;
    const float cf = w_conf[b * kN + n];
    const float fk = w_valid[b * kN + n] ? cf : kNegInf;
    unsigned u = __float_as_uint(fk);
    u = (u & 0x80000000u) ? ~u : (u | 0x80000000u);  // order-preserving map
    skey[n] = ((unsigned long long)u << 32) | (unsigned)(~(unsigned)n);
  }

  for (unsigned k = 2; k <= (unsigned)kN; k <<= 1) {
    for (unsigned j = k >> 1; j > 0; j >>= 1) {
      __syncthreads();
      for (int s = 0; s < 4; ++s) {
        const unsigned i = (unsigned)tid + (unsigned)s * 1024u;
        const unsigned ixj = i ^ j;
        if (ixj > i) {
          const unsigned long long a  = skey[i];
          const unsigned long long c2 = skey[ixj];
          const bool up = ((i & k) == 0u);
          if ((a > c2) == up) { skey[i] = c2; skey[ixj] = a; }
        }
      }
    }
  }
  __syncthreads();

  for (int s = 0; s < 4; ++s) {
    const int pos = tid + s * 1024;
    const int r = kN - 1 - pos;  // descending rank
    const unsigned low = (unsigned)(skey[pos] & 0xffffffffull);
    const int idx = (int)(~low);

    const float4 bx = ((const float4*)w_box)[b * kN + idx];
    const int c = w_cls[b * kN + idx];
    const float off = (float)c * kMaxWh;  // class-aware NMS coordinate offset

    ((float4*)w_sbox)[b * kN + r] = bx;
    ((float4*)w_soff)[b * kN + r] =
        make_float4(bx.x + off, bx.y + off, bx.z + off, bx.w + off);
    w_sconf[b * kN + r]  = w_conf[b * kN + idx];
    w_scls[b * kN + r]   = (float)c;
    w_svalid[b * kN + r] = w_valid[b * kN + idx];
  }
}

// ============================================================================
// Kernel 3: per-batch greedy NMS (exactly equivalent to the reference scan:
// row i only suppresses if keep[i] still set) + padded output write.
// Offset boxes (64 KB) staged into LDS via CDNA5 async global->LDS DMA.
// keep bitmap (128 words) lives in LDS, cleared with ds_and_b32 atomics.
// Dynamic LDS: 4096*16 + 128*4 = 66048 B (CDNA5 allows 320 KB/workgroup).
// ============================================================================
__global__ __launch_bounds__(1024) void k_nms(
    const float* __restrict__ w_soff, const float* __restrict__ w_sbox,
    const float* __restrict__ w_sconf, const float* __restrict__ w_scls,
    const unsigned* __restrict__ w_svalid, float* __restrict__ out) {
  extern __shared__ unsigned char smem[];
  float4*   offb  = (float4*)smem;
  unsigned* keepw = (unsigned*)(smem + (size_t)kN * sizeof(float4));

  const int b = blockIdx.x;
  const int tid = threadIdx.x;
  const float4* src = (const float4*)w_soff + (size_t)b * kN;

#ifdef OVD_ASYNC_LDS
  for (int p = tid; p < kN; p += 1024) {
    __builtin_amdgcn_global_load_async_to_lds_b128(
        (ovd_gptr_b128)(src + p),   // global source (AS1, non-const int4*)
        (ovd_lptr_b128)(offb + p),  // LDS destination (AS3 int4*)
        /*offset=*/0, /*cpol=*/0);
  }
#ifdef OVD_HAVE_WAIT_ASYNC_BUILTIN
  __builtin_amdgcn_s_wait_asynccnt(0);
#else
  asm volatile("s_wait_asynccnt 0" ::: "memory");
#endif
#else
  for (int p = tid; p < kN; p += 1024) offb[p] = src[p];
#endif

  // keep bitmap init = sorted validity
  if (tid < kN / 32) {
    const unsigned* sv = w_svalid + (size_t)b * kN + tid * 32;
    unsigned wv = 0;
    for (int q = 0; q < 32; ++q) wv |= (sv[q] & 1u) << q;
    keepw[tid] = wv;
  }
  __syncthreads();

  for (int i = 0; i < kN; ++i) {
    const bool ki = (keepw[i >> 5] >> (i & 31)) & 1u;
    if (ki) {
      const float4 bi = offb[i];
      const float ai =
          fmaxf(bi.z - bi.x, 0.0f) * fmaxf(bi.w - bi.y, 0.0f);
      for (int j = i + 1 + tid; j < kN; j += 1024) {
        if (!((keepw[j >> 5] >> (j & 31)) & 1u)) continue;  // monotone, race-safe
        const float4 bj = offb[j];
        const float aj =
            fmaxf(bj.z - bj.x, 0.0f) * fmaxf(bj.w - bj.y, 0.0f);
        const float lx = fmaxf(bi.x, bj.x);
        const float ly = fmaxf(bi.y, bj.y);
        const float rx = fminf(bi.z, bj.z);
        const float ry = fminf(bi.w, bj.w);
        const float iw = fmaxf(rx - lx, 0.0f);
        const float ih = fmaxf(ry - ly, 0.0f);
        const float inter = iw * ih;
        const float uni = ai + aj - inter;
        const float iou = inter / (uni + 1e-7f);
        if (iou > kIouThr) atomicAnd(&keepw[j >> 5], ~(1u << (j & 31)));
      }
    }
    __syncthreads();
  }

  // padded detections [B,N,6] then keep mask [B,N] (as 0/1 float)
  float* outm = out + (size_t)kB * kN * 6;
  for (int p = tid; p < kN; p += 1024) {
    const bool kp = (keepw[p >> 5] >> (p & 31)) & 1u;
    const float4 bx = ((const float4*)w_sbox)[(size_t)b * kN + p];
    const size_t base = ((size_t)(b * kN + p)) * 6;
    out[base + 0] = kp ? bx.x : 0.0f;
    out[base + 1] = kp ? bx.y : 0.0f;
    out[base + 2] = kp ? bx.z : 0.0f;
    out[base + 3] = kp ? bx.w : 0.0f;
    out[base + 4] = kp ? w_sconf[(size_t)b * kN + p] : 0.0f;
    out[base + 5] = kp ? w_scls[(size_t)b * kN + p] : 0.0f;
    outm[(size_t)b * kN + p] = kp ? 1.0f : 0.0f;
  }
}

// ============================================================================
// Launcher
// ============================================================================
extern "C" void kernel_launch(void* const* d_in, const int* in_sizes, int n_in,
                              void* d_out, int out_size, void* d_ws, size_t ws_size,
                              hipStream_t stream) {
  (void)in_sizes; (void)n_in; (void)out_size; (void)ws_size;

  const float* prop  = (const float*)d_in[0];  // [B,N,4]
  const float* preds = (const float*)d_in[1];  // [B,N,C]
  const float* obj   = (const float*)d_in[2];  // [B,N]
  float* out = (float*)d_out;                  // [B,N,6] ++ [B,N]

  // Workspace layout (all 16-byte aligned slices): ~4.5 MB total
  float*    ws       = (float*)d_ws;
  float*    w_box    = ws;                                  // B*N*4
  float*    w_conf   = w_box + (size_t)4 * kB * kN;         // B*N
  int*      w_cls    = (int*)(w_conf + (size_t)kB * kN);    // B*N
  unsigned* w_valid  = (unsigned*)(w_cls + (size_t)kB * kN);// B*N
  float*    w_sbox   = (float*)(w_valid + (size_t)kB * kN); // B*N*4
  float*    w_soff   = w_sbox + (size_t)4 * kB * kN;        // B*N*4
  float*    w_sconf  = w_soff + (size_t)4 * kB * kN;        // B*N
  float*    w_scls   = w_sconf + (size_t)kB * kN;           // B*N
  unsigned* w_svalid = (unsigned*)(w_scls + (size_t)kB * kN);// B*N

  k_prep<<<(kB * kN) / 256, 256, 0, stream>>>(
      prop, preds, obj, w_box, w_conf, w_cls, w_valid);

  k_sort<<<kB, 1024, 0, stream>>>(
      w_box, w_conf, w_cls, w_valid,
      w_sbox, w_soff, w_sconf, w_scls, w_svalid);

  const size_t nms_lds = (size_t)kN * sizeof(float4) + (kN / 32) * sizeof(unsigned);
  k_nms<<<kB, 1024, nms_lds, stream>>>(
      w_soff, w_sbox, w_sconf, w_scls, w_svalid, out);
}